// GAT_24137716203812
// MI455X (gfx1250) — compile-verified
//
#include <hip/hip_runtime.h>
#include <math.h>

// ---------------- problem constants (match reference) ----------------
#define N_NODES 100000
#define N_EDGES 1600000
#define F_IN    128
#define HID     64
#define OUT_C   2
#define NEG_SLOPE 0.2f

typedef float v2f __attribute__((ext_vector_type(2)));
typedef float v8f __attribute__((ext_vector_type(8)));

// float atomic max via signed/unsigned integer atomics (valid with -inf init)
__device__ __forceinline__ void atomicMaxF(float* addr, float v) {
    if (v >= 0.0f) atomicMax((int*)addr, __float_as_int(v));
    else           atomicMin((unsigned int*)addr, __float_as_uint(v));
}

// =====================================================================
// Layer-1 fused GEMM: [N,128] x [128, 64|64|64] -> hs, hd, hlin  (fp32 WMMA)
// One wave computes one 16x16 output tile; K-loop of 32 x v_wmma_f32_16x16x4_f32.
// A layout (16x4 f32): lane&15 = row M; VGPR0/1 = K {0,1} (lanes 0-15) or {2,3} (lanes 16-31)
// C/D layout: value(v,lane) = C[M = v + 8*(lane>>4)][Ncol = lane&15]
// =====================================================================
__global__ void gat_gemm1(const float* __restrict__ x,
                          const float* __restrict__ Wsrc,
                          const float* __restrict__ Wdst,
                          const float* __restrict__ Wlin,
                          float* __restrict__ hs,
                          float* __restrict__ hd,
                          float* __restrict__ hlin)
{
    const int wave  = (blockIdx.x << 3) + (threadIdx.x >> 5);   // 8 waves / 256-thread block
    const int lane  = threadIdx.x & 31;
    const int mtile = wave / 12;          // 6250 M tiles of 16 rows
    const int ntile = wave % 12;          // 12 N tiles of 16 cols (3 matrices x 4 tiles)
    if (mtile >= (N_NODES >> 4)) return;  // wave-uniform

    const int sel = ntile >> 2;
    const float* __restrict__ B   = (sel == 0) ? Wsrc : (sel == 1) ? Wdst : Wlin;
    float*       __restrict__ Out = (sel == 0) ? hs   : (sel == 1) ? hd   : hlin;

    const int col   = ((ntile & 3) << 4) + (lane & 15);   // 0..63 within selected W
    const int khalf = (lane >> 4) << 1;                   // 0 or 2
    const float* __restrict__ arow = x + (size_t)((mtile << 4) + (lane & 15)) * F_IN;

    v8f acc = {0.f, 0.f, 0.f, 0.f, 0.f, 0.f, 0.f, 0.f};
    #pragma unroll 8
    for (int k = 0; k < F_IN; k += 4) {
        const int ka = k + khalf;
        v2f a; a.x = arow[ka];            a.y = arow[ka + 1];
        v2f b; b.x = B[ka * HID + col];   b.y = B[(ka + 1) * HID + col];
        acc = __builtin_amdgcn_wmma_f32_16x16x4_f32(false, a, false, b,
                                                    (short)0, acc, false, false);
    }
    const int orow0 = (mtile << 4) + ((lane >> 4) << 3);
    #pragma unroll
    for (int v = 0; v < 8; ++v)
        Out[(size_t)(orow0 + v) * HID + col] = acc[v];
}

// =====================================================================
// Layer-2 GEMM: [N,64] x [64, 2|2|2] -> hs2, hd2, hlin2 (6 valid cols, padded to 16)
// =====================================================================
__global__ void gat_gemm2(const float* __restrict__ h,
                          const float* __restrict__ W2s,
                          const float* __restrict__ W2d,
                          const float* __restrict__ Wl2,
                          float* __restrict__ hs2,
                          float* __restrict__ hd2,
                          float* __restrict__ hlin2)
{
    const int wave  = (blockIdx.x << 3) + (threadIdx.x >> 5);
    const int lane  = threadIdx.x & 31;
    const int mtile = wave;
    if (mtile >= (N_NODES >> 4)) return;  // wave-uniform

    const int col   = lane & 15;
    const int khalf = (lane >> 4) << 1;
    const float* __restrict__ arow = h + (size_t)((mtile << 4) + col) * HID;

    v8f acc = {0.f, 0.f, 0.f, 0.f, 0.f, 0.f, 0.f, 0.f};
    #pragma unroll 4
    for (int k = 0; k < HID; k += 4) {
        const int ka = k + khalf;
        v2f a; a.x = arow[ka]; a.y = arow[ka + 1];
        float b0 = 0.f, b1 = 0.f;
        if (col < 2)      { b0 = W2s[ka * OUT_C + col];     b1 = W2s[(ka + 1) * OUT_C + col];     }
        else if (col < 4) { b0 = W2d[ka * OUT_C + col - 2]; b1 = W2d[(ka + 1) * OUT_C + col - 2]; }
        else if (col < 6) { b0 = Wl2[ka * OUT_C + col - 4]; b1 = Wl2[(ka + 1) * OUT_C + col - 4]; }
        v2f b; b.x = b0; b.y = b1;
        acc = __builtin_amdgcn_wmma_f32_16x16x4_f32(false, a, false, b,
                                                    (short)0, acc, false, false);
    }
    const int orow0 = (mtile << 4) + ((lane >> 4) << 3);
    #pragma unroll
    for (int v = 0; v < 8; ++v) {
        const int r = orow0 + v;
        if (col < 2)      hs2  [(size_t)r * OUT_C + col    ] = acc[v];
        else if (col < 4) hd2  [(size_t)r * OUT_C + col - 2] = acc[v];
        else if (col < 6) hlin2[(size_t)r * OUT_C + col - 4] = acc[v];
    }
}

// =====================================================================
// Layer-1 node prep: es = hs . a_s, ed = hd . a_d (wave-per-node shuffle reduce),
// plus init of emax/esum/agg.
// =====================================================================
__global__ void node1_prep(const float* __restrict__ hs, const float* __restrict__ hd,
                           const float* __restrict__ as, const float* __restrict__ ad,
                           float* __restrict__ es, float* __restrict__ ed,
                           float* __restrict__ emax, float* __restrict__ esum,
                           float* __restrict__ agg)
{
    const int node = (blockIdx.x << 3) + (threadIdx.x >> 5);
    const int lane = threadIdx.x & 31;
    if (node >= N_NODES) return;
    const size_t base = (size_t)node * HID;
    float s = hs[base + lane] * as[lane] + hs[base + lane + 32] * as[lane + 32];
    float d = hd[base + lane] * ad[lane] + hd[base + lane + 32] * ad[lane + 32];
    #pragma unroll
    for (int off = 16; off > 0; off >>= 1) {
        s += __shfl_xor(s, off, 32);
        d += __shfl_xor(d, off, 32);
    }
    agg[base + lane]      = 0.f;
    agg[base + lane + 32] = 0.f;
    if (lane == 0) {
        es[node] = s; ed[node] = d;
        emax[node] = -INFINITY; esum[node] = 0.f;
    }
}

// Layer-2 node prep (C=2): thread per node
__global__ void node2_prep(const float* __restrict__ hs2, const float* __restrict__ hd2,
                           const float* __restrict__ a2s, const float* __restrict__ a2d,
                           float* __restrict__ es, float* __restrict__ ed,
                           float* __restrict__ emax, float* __restrict__ esum,
                           float* __restrict__ agg)
{
    const int n = blockIdx.x * blockDim.x + threadIdx.x;
    if (n >= N_NODES) return;
    es[n] = hs2[n * 2] * a2s[0] + hs2[n * 2 + 1] * a2s[1];
    ed[n] = hd2[n * 2] * a2d[0] + hd2[n * 2 + 1] * a2d[1];
    emax[n] = -INFINITY; esum[n] = 0.f;
    agg[n * 2] = 0.f; agg[n * 2 + 1] = 0.f;
}

// Edge pass 1: logit + leaky relu + segment max (generic for both layers)
__global__ void edge_logit_max(const int* __restrict__ src, const int* __restrict__ dst,
                               const float* __restrict__ es, const float* __restrict__ ed,
                               float* __restrict__ ebuf, float* __restrict__ emax)
{
    const int e = blockIdx.x * blockDim.x + threadIdx.x;
    if (e >= N_EDGES) return;
    const int s = src[e], d = dst[e];
    float v = es[s] + ed[d];
    v = (v > 0.f) ? v : NEG_SLOPE * v;
    ebuf[e] = v;
    atomicMaxF(&emax[d], v);
}

// Edge pass 2: p = exp(e - max), segment sum (generic)
__global__ void edge_exp_sum(const int* __restrict__ dst,
                             float* __restrict__ ebuf,
                             const float* __restrict__ emax, float* __restrict__ esum)
{
    const int e = blockIdx.x * blockDim.x + threadIdx.x;
    if (e >= N_EDGES) return;
    const int d = dst[e];
    const float p = __expf(ebuf[e] - emax[d]);
    ebuf[e] = p;
    atomicAdd(&esum[d], p);
}

// Edge pass 3 (layer 1, C=64): wave per edge, alpha-weighted scatter of hs[src]
__global__ void edge_aggregate64(const int* __restrict__ src, const int* __restrict__ dst,
                                 const float* __restrict__ ebuf, const float* __restrict__ esum,
                                 const float* __restrict__ hs, float* __restrict__ agg)
{
    const int e    = (blockIdx.x << 3) + (threadIdx.x >> 5);
    const int lane = threadIdx.x & 31;
    if (e >= N_EDGES) return;
    const int s = src[e], d = dst[e];
    const float alpha = ebuf[e] / esum[d];
    const size_t sb = (size_t)s * HID, db = (size_t)d * HID;
    atomicAdd(&agg[db + lane],      hs[sb + lane]      * alpha);
    atomicAdd(&agg[db + lane + 32], hs[sb + lane + 32] * alpha);
}

// Edge pass 3 (layer 2, C=2): thread per edge
__global__ void edge_aggregate2(const int* __restrict__ src, const int* __restrict__ dst,
                                const float* __restrict__ ebuf, const float* __restrict__ esum,
                                const float* __restrict__ hs2, float* __restrict__ agg2)
{
    const int e = blockIdx.x * blockDim.x + threadIdx.x;
    if (e >= N_EDGES) return;
    const int s = src[e], d = dst[e];
    const float alpha = ebuf[e] / esum[d];
    atomicAdd(&agg2[d * 2],     hs2[s * 2]     * alpha);
    atomicAdd(&agg2[d * 2 + 1], hs2[s * 2 + 1] * alpha);
}

// h = relu(agg + b1 + hlin + bl1)
__global__ void fuse_relu(const float* __restrict__ agg, const float* __restrict__ hlin,
                          const float* __restrict__ b1, const float* __restrict__ bl1,
                          float* __restrict__ h)
{
    const int i = blockIdx.x * blockDim.x + threadIdx.x;
    if (i >= N_NODES * HID) return;
    const int c = i & (HID - 1);
    const float v = agg[i] + b1[c] + hlin[i] + bl1[c];
    h[i] = (v > 0.f) ? v : 0.f;
}

// out = sigmoid(agg2 + b2 + hlin2 + bl2)
__global__ void fuse_sigmoid(const float* __restrict__ agg2, const float* __restrict__ hlin2,
                             const float* __restrict__ b2, const float* __restrict__ bl2,
                             float* __restrict__ out)
{
    const int i = blockIdx.x * blockDim.x + threadIdx.x;
    if (i >= N_NODES * OUT_C) return;
    const int c = i & (OUT_C - 1);
    const float v = agg2[i] + b2[c] + hlin2[i] + bl2[c];
    out[i] = 1.0f / (1.0f + __expf(-v));
}

// =====================================================================
extern "C" void kernel_launch(void* const* d_in, const int* in_sizes, int n_in,
                              void* d_out, int out_size, void* d_ws, size_t ws_size,
                              hipStream_t stream)
{
    const float* x   = (const float*)d_in[0];
    const int*   ei  = (const int*)  d_in[1];
    const float* W1s = (const float*)d_in[2];
    const float* W1d = (const float*)d_in[3];
    const float* a1s = (const float*)d_in[4];
    const float* a1d = (const float*)d_in[5];
    const float* b1  = (const float*)d_in[6];
    const float* Wl1 = (const float*)d_in[7];
    const float* bl1 = (const float*)d_in[8];
    const float* W2s = (const float*)d_in[9];
    const float* W2d = (const float*)d_in[10];
    const float* a2s = (const float*)d_in[11];
    const float* a2d = (const float*)d_in[12];
    const float* b2  = (const float*)d_in[13];
    const float* Wl2 = (const float*)d_in[14];
    const float* bl2 = (const float*)d_in[15];

    const int* src = ei;            // edge_index[0,:]
    const int* dst = ei + N_EDGES;  // edge_index[1,:]

    // ---- workspace layout (floats) ~115 MB ----
    float* w = (float*)d_ws;
    float* hs1   = w; w += (size_t)N_NODES * HID;
    float* hd1   = w; w += (size_t)N_NODES * HID;   // reused as h after layer 1
    float* hlin1 = w; w += (size_t)N_NODES * HID;
    float* agg1  = w; w += (size_t)N_NODES * HID;
    float* es1   = w; w += N_NODES;
    float* ed1   = w; w += N_NODES;
    float* emax1 = w; w += N_NODES;
    float* esum1 = w; w += N_NODES;
    float* ebuf  = w; w += N_EDGES;                 // shared by both layers
    float* hs2   = w; w += (size_t)N_NODES * OUT_C;
    float* hd2   = w; w += (size_t)N_NODES * OUT_C;
    float* hlin2 = w; w += (size_t)N_NODES * OUT_C;
    float* agg2  = w; w += (size_t)N_NODES * OUT_C;
    float* es2   = w; w += N_NODES;
    float* ed2   = w; w += N_NODES;
    float* emax2 = w; w += N_NODES;
    float* esum2 = w; w += N_NODES;
    float* h     = hd1;  // alias: hd1 dead after node1_prep

    const int edgeBlocks  = (N_EDGES + 255) / 256;        // thread-per-edge kernels
    const int edgeWaveBlk = N_EDGES / 8;                  // wave-per-edge (1.6M % 8 == 0)

    // ---- layer 1 ----
    // 6250 M-tiles * 12 N-tiles = 75000 waves, 8 waves/block
    gat_gemm1<<<75000 / 8, 256, 0, stream>>>(x, W1s, W1d, Wl1, hs1, hd1, hlin1);
    node1_prep<<<N_NODES / 8, 256, 0, stream>>>(hs1, hd1, a1s, a1d,
                                                es1, ed1, emax1, esum1, agg1);
    edge_logit_max<<<edgeBlocks, 256, 0, stream>>>(src, dst, es1, ed1, ebuf, emax1);
    edge_exp_sum<<<edgeBlocks, 256, 0, stream>>>(dst, ebuf, emax1, esum1);
    edge_aggregate64<<<edgeWaveBlk, 256, 0, stream>>>(src, dst, ebuf, esum1, hs1, agg1);
    fuse_relu<<<(N_NODES * HID) / 256, 256, 0, stream>>>(agg1, hlin1, b1, bl1, h);

    // ---- layer 2 ----
    gat_gemm2<<<(6250 + 7) / 8, 256, 0, stream>>>(h, W2s, W2d, Wl2, hs2, hd2, hlin2);
    node2_prep<<<(N_NODES + 255) / 256, 256, 0, stream>>>(hs2, hd2, a2s, a2d,
                                                          es2, ed2, emax2, esum2, agg2);
    edge_logit_max<<<edgeBlocks, 256, 0, stream>>>(src, dst, es2, ed2, ebuf, emax2);
    edge_exp_sum<<<edgeBlocks, 256, 0, stream>>>(dst, ebuf, emax2, esum2);
    edge_aggregate2<<<edgeBlocks, 256, 0, stream>>>(src, dst, ebuf, esum2, hs2, agg2);
    fuse_sigmoid<<<(N_NODES * OUT_C + 255) / 256, 256, 0, stream>>>(agg2, hlin2, b2, bl2,
                                                                    (float*)d_out);
}